// Gene2Gauss_mv_53772990545981
// MI455X (gfx1250) — compile-verified
//
#include <hip/hip_runtime.h>
#include <math.h>

#define N_NODES 50000
#define D_DIM   256
#define H_DIM   128
#define L_DIM   64
#define V_VIEWS 3
#define E_EDGES 1600000
#define P_PAIRS 10000
#define VH      (V_VIEWS * H_DIM)   /* 384 */

typedef __attribute__((ext_vector_type(2))) float v2f;
typedef __attribute__((ext_vector_type(8))) float v8f;

// ---------------------------------------------------------------------------
// K0: zero the accumulation buffer (concat) and the scalar output. float4.
// ---------------------------------------------------------------------------
__global__ void zero_kernel(float4* __restrict__ concat4, float* __restrict__ out) {
    size_t i = (size_t)blockIdx.x * blockDim.x + threadIdx.x;
    const size_t total4 = (size_t)N_NODES * VH / 4;
    if (i < total4) concat4[i] = make_float4(0.f, 0.f, 0.f, 0.f);
    if (i == 0) out[0] = 0.0f;
}

// ---------------------------------------------------------------------------
// K1: XW[v] = X @ W[v]   (fp32 WMMA 16x16, K-step 4)
// One wave per 16x32 output strip: two accumulators share each A-fragment.
// A-frag layout (32-bit A 16x4): lanes 0-15 -> K=0,1 ; lanes 16-31 -> K=2,3
// C/D layout: VGPR r -> M = r (+8 for lanes 16-31), N = lane & 15
// ---------------------------------------------------------------------------
__global__ void gemm_xw_kernel(const float* __restrict__ X,
                               const float* __restrict__ W,
                               float* __restrict__ XW) {
    const int lane = threadIdx.x & 31;
    const int wave = threadIdx.x >> 5;
    const int stripsPerView = (N_NODES / 16) * (H_DIM / 32);   // 3125*4 = 12500
    int strip = blockIdx.x * (blockDim.x >> 5) + wave;
    if (strip >= V_VIEWS * stripsPerView) return;              // whole-wave uniform

    const int v   = strip / stripsPerView;
    const int rem = strip % stripsPerView;
    const int mt  = rem / (H_DIM / 32);
    const int nt  = rem % (H_DIM / 32);
    const int col0 = nt * 32;

    const float* __restrict__ Xrow = X + (size_t)(mt * 16) * D_DIM;
    const float* __restrict__ Wv   = W + (size_t)v * D_DIM * H_DIM + col0;

    const int l  = lane & 15;
    const int kh = (lane >> 4) << 1;     // 0 or 2

    v8f acc0 = {};
    v8f acc1 = {};
    for (int k = 0; k < D_DIM; k += 4) {
        v2f a, b0, b1;
        a.x = Xrow[(size_t)l * D_DIM + k + kh];
        a.y = Xrow[(size_t)l * D_DIM + k + kh + 1];
        b0.x = Wv[(size_t)(k + kh) * H_DIM + l];
        b0.y = Wv[(size_t)(k + kh + 1) * H_DIM + l];
        b1.x = Wv[(size_t)(k + kh) * H_DIM + 16 + l];
        b1.y = Wv[(size_t)(k + kh + 1) * H_DIM + 16 + l];
        acc0 = __builtin_amdgcn_wmma_f32_16x16x4_f32(
            false, a, false, b0, (short)0, acc0, false, false);
        acc1 = __builtin_amdgcn_wmma_f32_16x16x4_f32(
            false, a, false, b1, (short)0, acc1, false, false);
    }

    float* __restrict__ Drow =
        XW + ((size_t)v * N_NODES + (size_t)mt * 16) * H_DIM + col0;
    const int mbase = (lane >> 4) << 3;
    for (int r = 0; r < 8; ++r) {
        Drow[(size_t)(mbase + r) * H_DIM + l]      = acc0[r];
        Drow[(size_t)(mbase + r) * H_DIM + 16 + l] = acc1[r];
    }
}

// ---------------------------------------------------------------------------
// K2: edge scatter. Grid-stride loop, one wave32 per edge iteration:
// 32 lanes x float4 = the full 512B XW[src] row in one coalesced b128 load,
// then 4 fp32 atomics/lane into concat[dst, v*H + :].
// concat (76.8 MB) fits in the 192 MB L2 -> atomics resolve on-chip.
// ---------------------------------------------------------------------------
__global__ void scatter_kernel(const int* __restrict__ edge_idx,
                               const float* __restrict__ edge_vals,
                               const float* __restrict__ XW,
                               float* __restrict__ concat) {
    const int lane = threadIdx.x & 31;
    const size_t wavesTotal = (size_t)gridDim.x * (blockDim.x >> 5);
    size_t wid = (size_t)blockIdx.x * (blockDim.x >> 5) + (threadIdx.x >> 5);
    const size_t totalEdges = (size_t)V_VIEWS * E_EDGES;

    for (size_t ei = wid; ei < totalEdges; ei += wavesTotal) {
        const int v   = (int)(ei / E_EDGES);
        const int src = edge_idx[ei * 2 + 0];
        const int dst = edge_idx[ei * 2 + 1];
        const float val = edge_vals[ei];

        const float4* __restrict__ xrow =
            (const float4*)(XW + ((size_t)v * N_NODES + src) * H_DIM);
        float4 x = xrow[lane];                       // lane*16B, 32 lanes = 512B

        float* __restrict__ crow =
            concat + (size_t)dst * VH + (size_t)v * H_DIM + lane * 4;
        atomicAdd(crow + 0, val * x.x);
        atomicAdd(crow + 1, val * x.y);
        atomicAdd(crow + 2, val * x.z);
        atomicAdd(crow + 3, val * x.w);
    }
}

// ---------------------------------------------------------------------------
// K3: concat = relu(concat + b[v,h])   (float4)
// ---------------------------------------------------------------------------
__global__ void bias_relu_kernel(float4* __restrict__ concat4,
                                 const float4* __restrict__ b4) {
    size_t g = (size_t)blockIdx.x * blockDim.x + threadIdx.x;
    const size_t total4 = (size_t)N_NODES * VH / 4;
    if (g < total4) {
        float4 c = concat4[g];
        float4 bb = b4[g % (VH / 4)];               // VH divisible by 4
        c.x = fmaxf(c.x + bb.x, 0.0f);
        c.y = fmaxf(c.y + bb.y, 0.0f);
        c.z = fmaxf(c.z + bb.z, 0.0f);
        c.w = fmaxf(c.w + bb.w, 0.0f);
        concat4[g] = c;
    }
}

// ---------------------------------------------------------------------------
// K4: mu = concat @ Wmu + bmu ; sigma = elu(concat @ Wsig + bsig) + 1 + 1e-14
// Fused: shared A-fragments feed two fp32 WMMA accumulators.
// 4 waves / block, wave = L-column tile (L/16 = 4), block = M tile.
// ---------------------------------------------------------------------------
__global__ void gemm_musig_kernel(const float* __restrict__ concat,
                                  const float* __restrict__ Wmu,
                                  const float* __restrict__ bmu,
                                  const float* __restrict__ Wsig,
                                  const float* __restrict__ bsig,
                                  float* __restrict__ mu,
                                  float* __restrict__ sigma) {
    const int lane = threadIdx.x & 31;
    const int nt   = threadIdx.x >> 5;     // 0..3
    const int mt   = blockIdx.x;           // 0..3124

    const float* __restrict__ Arow = concat + (size_t)(mt * 16) * VH;
    const int l  = lane & 15;
    const int kh = (lane >> 4) << 1;

    v8f accm = {};
    v8f accs = {};
    for (int k = 0; k < VH; k += 4) {
        v2f a;
        a.x = Arow[(size_t)l * VH + k + kh];
        a.y = Arow[(size_t)l * VH + k + kh + 1];
        v2f bm, bs;
        bm.x = Wmu[(size_t)(k + kh) * L_DIM + nt * 16 + l];
        bm.y = Wmu[(size_t)(k + kh + 1) * L_DIM + nt * 16 + l];
        bs.x = Wsig[(size_t)(k + kh) * L_DIM + nt * 16 + l];
        bs.y = Wsig[(size_t)(k + kh + 1) * L_DIM + nt * 16 + l];
        accm = __builtin_amdgcn_wmma_f32_16x16x4_f32(
            false, a, false, bm, (short)0, accm, false, false);
        accs = __builtin_amdgcn_wmma_f32_16x16x4_f32(
            false, a, false, bs, (short)0, accs, false, false);
    }

    const int col   = nt * 16 + l;
    const int mbase = (lane >> 4) << 3;
    const float bm_c = bmu[col];
    const float bs_c = bsig[col];
    for (int r = 0; r < 8; ++r) {
        size_t row = (size_t)mt * 16 + mbase + r;
        mu[row * L_DIM + col] = accm[r] + bm_c;
        float z = accs[r] + bs_c;
        // elu(z) + 1 + 1e-14 :  z>0 -> z+1 ; else exp(z)
        float s = (z > 0.0f ? z + 1.0f : __expf(z)) + 1e-14f;
        sigma[row * L_DIM + col] = s;
    }
}

// ---------------------------------------------------------------------------
// K5: KL energy. One wave32 per pair p; lanes cover L=64 (2 elems each),
// shuffle-reduce, then a single atomic add of (e_pos^2 + exp(-e_neg)) / P.
// ---------------------------------------------------------------------------
__global__ void energy_kernel(const float* __restrict__ mu,
                              const float* __restrict__ sigma,
                              const int* __restrict__ pos,
                              const int* __restrict__ neg,
                              float* __restrict__ out) {
    const int lane = threadIdx.x & 31;
    const int wave = threadIdx.x >> 5;
    const int p = blockIdx.x * (blockDim.x >> 5) + wave;
    if (p >= P_PAIRS) return;

    float eng[2];
    for (int which = 0; which < 2; ++which) {
        const int* __restrict__ pr = (which == 0) ? pos : neg;
        const int i = pr[p * 2 + 0];
        const int j = pr[p * 2 + 1];
        const float* __restrict__ mi = mu    + (size_t)i * L_DIM;
        const float* __restrict__ mj = mu    + (size_t)j * L_DIM;
        const float* __restrict__ si = sigma + (size_t)i * L_DIM;
        const float* __restrict__ sj = sigma + (size_t)j * L_DIM;

        float tr = 0.0f, ld = 0.0f, md = 0.0f;
        for (int t = lane; t < L_DIM; t += 32) {
            float siv = si[t], sjv = sj[t];
            float ratio = sjv / siv;
            tr += ratio;
            ld += logf(ratio + 1e-14f);
            float d = mi[t] - mj[t];
            md += d * d / siv;
        }
        for (int off = 16; off > 0; off >>= 1) {
            tr += __shfl_down(tr, off, 32);
            ld += __shfl_down(ld, off, 32);
            md += __shfl_down(md, off, 32);
        }
        eng[which] = 0.5f * (tr + md - (float)L_DIM - ld);
    }

    if (lane == 0) {
        float e = eng[0] * eng[0] + __expf(-eng[1]);
        atomicAdd(out, e * (1.0f / (float)P_PAIRS));
    }
}

// ---------------------------------------------------------------------------
// Launch
// ---------------------------------------------------------------------------
extern "C" void kernel_launch(void* const* d_in, const int* in_sizes, int n_in,
                              void* d_out, int out_size, void* d_ws, size_t ws_size,
                              hipStream_t stream) {
    (void)in_sizes; (void)n_in; (void)out_size; (void)ws_size;

    const float* X         = (const float*)d_in[0];
    const int*   edge_idx  = (const int*)  d_in[1];
    const float* edge_vals = (const float*)d_in[2];
    const float* W         = (const float*)d_in[3];
    const float* b         = (const float*)d_in[4];
    const float* Wmu       = (const float*)d_in[5];
    const float* bmu       = (const float*)d_in[6];
    const float* Wsig      = (const float*)d_in[7];
    const float* bsig      = (const float*)d_in[8];
    const int*   pos       = (const int*)  d_in[9];
    const int*   neg       = (const int*)  d_in[10];
    float* out = (float*)d_out;

    // Workspace layout (floats):
    float* ws     = (float*)d_ws;
    float* concat = ws;                                         // N*VH   = 19.2M
    float* XW     = concat + (size_t)N_NODES * VH;              // V*N*H  = 19.2M
    float* muB    = XW + (size_t)V_VIEWS * N_NODES * H_DIM;     // N*L    =  3.2M
    float* sigB   = muB + (size_t)N_NODES * L_DIM;              // N*L    =  3.2M

    const size_t total4 = (size_t)N_NODES * VH / 4;             // 4.8M float4
    const int vecBlocks = (int)((total4 + 255) / 256);

    zero_kernel<<<vecBlocks, 256, 0, stream>>>((float4*)concat, out);

    // 37500 strips / 8 waves-per-block = 4688 blocks (guarded tail)
    gemm_xw_kernel<<<4688, 256, 0, stream>>>(X, W, XW);

    // grid-stride: 76800 waves over 4.8M edges
    scatter_kernel<<<9600, 256, 0, stream>>>(edge_idx, edge_vals, XW, concat);

    bias_relu_kernel<<<vecBlocks, 256, 0, stream>>>((float4*)concat, (const float4*)b);

    gemm_musig_kernel<<<N_NODES / 16, 128, 0, stream>>>(
        concat, Wmu, bmu, Wsig, bsig, muB, sigB);

    energy_kernel<<<(P_PAIRS + 7) / 8, 256, 0, stream>>>(muB, sigB, pos, neg, out);
}